// lct_fk_34205119545601
// MI455X (gfx1250) — compile-verified
//
#include <hip/hip_runtime.h>

// ---------------------------------------------------------------------------
// FK-migration (Stolt) NLOS pipeline for MI455X (gfx1250), wave32 + WMMA.
//
// Volume: 512 x 256 x 256 complex fp32. All FFT axis passes are fused
// LDS-tile passes; the 16-point DFT stages run on the matrix pipe via
// V_WMMA_F32_16X16X4_F32 (full fp32 precision; pipeline is HBM-bound at
// ~4 GB traffic -> ~170us floor @ 23.3 TB/s, so f32 WMMA throughput is
// far more than sufficient).
// ---------------------------------------------------------------------------

typedef float v2f __attribute__((ext_vector_type(2)));
typedef float v8f __attribute__((ext_vector_type(8)));

#define FK_PI 3.14159265358979323846f

static __device__ __forceinline__ v8f wmma4(v2f a, v2f b, v8f c) {
  // D(16x16) = A(16x4) * B(4x16) + C, fp32 WMMA.
  return __builtin_amdgcn_wmma_f32_16x16x4_f32(
      /*neg_a=*/false, a, /*neg_b=*/false, b,
      /*c_mod=*/(short)0, c, /*reuse_a=*/false, /*reuse_b=*/false);
}

// ---------------------------------------------------------------------------
// 256-point complex FFT executed by ONE wave (32 lanes), data in LDS.
// Factorization: x = a + 16b, k = c + 16d
//   V[a][b] = v[a+16b]
//   G = V * F16            (16x16x16 complex matmul -> 16 WMMA)
//   H[a][c] = G[a][c] * w256^{a c}
//   X[c+16d] = (F16 * H)[d][c]   (16 WMMA)
// sgn = -1 forward, +1 inverse (unnormalized).
// vr/vi: element i lives at vr[i*stride]. hr/hi: 16x16 scratch, stride 17.
// ---------------------------------------------------------------------------
static __device__ __forceinline__ void fft256_wave(
    float* vr, float* vi, int stride, float* hr, float* hi, float sgn) {
  const int lane = threadIdx.x & 31;
  const int m = lane & 15;      // A row / B-D column
  const int hi16 = lane >> 4;   // which half of the K striping this lane owns
  const float w16step = sgn * (2.0f * FK_PI / 16.0f);

  // ---- Stage 1: G = V * F16 ----
  v8f gr = {}, gi = {};
#pragma unroll
  for (int k = 0; k < 4; ++k) {
    const int kb = 4 * k + 2 * hi16;  // this lane's K rows: kb, kb+1
    v2f ar, ai, br, bi, nbi;
    ar.x = vr[(m + 16 * kb) * stride];
    ar.y = vr[(m + 16 * (kb + 1)) * stride];
    ai.x = vi[(m + 16 * kb) * stride];
    ai.y = vi[(m + 16 * (kb + 1)) * stride];
    float c0, s0, c1, s1;
    __sincosf(w16step * (float)(kb * m), &s0, &c0);
    __sincosf(w16step * (float)((kb + 1) * m), &s1, &c1);
    br.x = c0; br.y = c1; bi.x = s0; bi.y = s1;
    nbi.x = -s0; nbi.y = -s1;
    gr = wmma4(ar, br, gr);
    gr = wmma4(ai, nbi, gr);
    gi = wmma4(ar, bi, gi);
    gi = wmma4(ai, br, gi);
  }

  // ---- Twiddle H = G .* w256^{a c}, re-stripe via LDS scratch ----
  const float w256step = sgn * (2.0f * FK_PI / 256.0f);
#pragma unroll
  for (int r = 0; r < 8; ++r) {
    const int a = r + 8 * hi16;
    float ct, st;
    __sincosf(w256step * (float)(a * m), &st, &ct);
    const float grv = gr[r], giv = gi[r];
    hr[a * 17 + m] = grv * ct - giv * st;
    hi[a * 17 + m] = grv * st + giv * ct;
  }
  __builtin_amdgcn_wave_barrier();

  // ---- Stage 2: X = F16 * H ----
  v8f xr = {}, xi = {};
#pragma unroll
  for (int k = 0; k < 4; ++k) {
    const int kb = 4 * k + 2 * hi16;
    v2f ar, ai, br, bi, nbi;
    float c0, s0, c1, s1;
    __sincosf(w16step * (float)(m * kb), &s0, &c0);
    __sincosf(w16step * (float)(m * (kb + 1)), &s1, &c1);
    ar.x = c0; ar.y = c1; ai.x = s0; ai.y = s1;
    br.x = hr[kb * 17 + m];       br.y = hr[(kb + 1) * 17 + m];
    bi.x = hi[kb * 17 + m];       bi.y = hi[(kb + 1) * 17 + m];
    nbi.x = -bi.x; nbi.y = -bi.y;
    xr = wmma4(ar, br, xr);
    xr = wmma4(ai, nbi, xr);
    xi = wmma4(ar, bi, xi);
    xi = wmma4(ai, br, xi);
  }
  __builtin_amdgcn_wave_barrier();

  // D element (row d = r+8*hi16, col c = m) is X[c + 16 d]; write back in place.
#pragma unroll
  for (int r = 0; r < 8; ++r) {
    const int d = r + 8 * hi16;
    vr[(m + 16 * d) * stride] = xr[r];
    vi[(m + 16 * d) * stride] = xi[r];
  }
  __builtin_amdgcn_wave_barrier();
}

// 512-point FFT = radix-2 DIF split + two 256-point FFTs.
// After this: lds[0..255] = X[even], lds[256..511] = X[odd].
static __device__ __forceinline__ void fft512_wave(
    float* vr, float* vi, int stride, float* hr, float* hi, float sgn) {
  const int lane = threadIdx.x & 31;
  const float wstep = sgn * (2.0f * FK_PI / 512.0f);
#pragma unroll
  for (int j = 0; j < 8; ++j) {
    const int i = lane + 32 * j;
    const float ar = vr[i * stride], ai = vi[i * stride];
    const float br = vr[(i + 256) * stride], bi = vi[(i + 256) * stride];
    const float ur = ar + br, ui = ai + bi;
    const float dr = ar - br, di = ai - bi;
    float c, s;
    __sincosf(wstep * (float)i, &s, &c);
    vr[i * stride] = ur;
    vi[i * stride] = ui;
    vr[(i + 256) * stride] = dr * c - di * s;
    vi[(i + 256) * stride] = dr * s + di * c;
  }
  __builtin_amdgcn_wave_barrier();
  fft256_wave(vr, vi, stride, hr, hi, sgn);
  fft256_wave(vr + 256 * stride, vi + 256 * stride, stride, hr, hi, sgn);
}

// ---------------------------------------------------------------------------
// Kernel 1: fused pad + preprocess + forward FFT along x. One wave per line.
// feat: [256,128,128] f32. A: [512,256,256] complex.
// ---------------------------------------------------------------------------
__global__ void __launch_bounds__(256) k_fwd_x(
    const float* __restrict__ feat, const int* __restrict__ ptbe,
    const int* __restrict__ pten, float2* __restrict__ A) {
  __shared__ float svr[8][256], svi[8][256];
  __shared__ float shr[8][272], shi[8][272];
  const int wave = threadIdx.x >> 5, lane = threadIdx.x & 31;
  const long line = (long)blockIdx.x * 8 + wave;  // z*256 + y
  const int z = (int)(line >> 8);
  const int y = (int)(line & 255);
  const long base = line * 256;

  const int tbe = ptbe[0], ten = pten[0];
  const int tsrc = z - tbe;
  const bool nz = (z < 256) && (y < 128) && (z >= tbe) && (z < ten) &&
                  (tsrc >= 0) && (tsrc < 256);
  if (!nz) {
#pragma unroll
    for (int j = 0; j < 8; ++j) A[base + lane + 32 * j] = make_float2(0.f, 0.f);
    return;
  }

  float* vr = svr[wave];
  float* vi = svi[wave];
  const float gz = (float)z * (1.0f / 255.0f);
  const float g2 = gz * gz;
#pragma unroll
  for (int j = 0; j < 8; ++j) {
    const int i = lane + 32 * j;
    float val = 0.f;
    if (i < 128) {
      val = feat[((long)tsrc * 128 + y) * 128 + i] * g2;  // * gridz^2
      val = (val > 0.f) ? sqrtf(val) : 0.f;               // relu + sqrt
    }
    vr[i] = val;
    vi[i] = 0.f;
  }
  __builtin_amdgcn_wave_barrier();
  fft256_wave(vr, vi, 1, shr[wave], shi[wave], -1.0f);
#pragma unroll
  for (int j = 0; j < 8; ++j) {
    const int i = lane + 32 * j;
    A[base + i] = make_float2(vr[i], vi[i]);
  }
}

// Generic complex FFT along x (used for the inverse pass). One wave per line.
__global__ void __launch_bounds__(256) k_pass_x(float2* __restrict__ buf,
                                                float sgn) {
  __shared__ float svr[8][256], svi[8][256];
  __shared__ float shr[8][272], shi[8][272];
  const int wave = threadIdx.x >> 5, lane = threadIdx.x & 31;
  const long base = ((long)blockIdx.x * 8 + wave) * 256;
  float* vr = svr[wave];
  float* vi = svi[wave];
#pragma unroll
  for (int j = 0; j < 8; ++j) {
    const int i = lane + 32 * j;
    const float2 v = buf[base + i];
    vr[i] = v.x;
    vi[i] = v.y;
  }
  __builtin_amdgcn_wave_barrier();
  fft256_wave(vr, vi, 1, shr[wave], shi[wave], sgn);
#pragma unroll
  for (int j = 0; j < 8; ++j) {
    const int i = lane + 32 * j;
    buf[base + i] = make_float2(vr[i], vi[i]);
  }
}

// FFT along y: LDS tile of 256(y) x 16(x) at fixed z; 8 waves, 2 columns each.
__global__ void __launch_bounds__(256) k_pass_y(float2* __restrict__ buf,
                                                float sgn) {
  __shared__ float tr[256 * 17], ti[256 * 17];
  __shared__ float shr[8][272], shi[8][272];
  const int wave = threadIdx.x >> 5;
  const int z = blockIdx.x >> 4;
  const int xbase = (blockIdx.x & 15) * 16;
  const long zbase = (long)z * 65536;  // 256*256

#pragma unroll
  for (int j = 0; j < 16; ++j) {
    const int e = threadIdx.x + 256 * j;  // 0..4095
    const int y = e >> 4, xi = e & 15;
    const float2 v = buf[zbase + (long)y * 256 + xbase + xi];
    tr[y * 17 + xi] = v.x;
    ti[y * 17 + xi] = v.y;
  }
  __syncthreads();
#pragma unroll
  for (int c = 0; c < 2; ++c) {
    const int col = wave * 2 + c;
    fft256_wave(tr + col, ti + col, 17, shr[wave], shi[wave], sgn);
  }
  __syncthreads();
#pragma unroll
  for (int j = 0; j < 16; ++j) {
    const int e = threadIdx.x + 256 * j;
    const int y = e >> 4, xi = e & 15;
    buf[zbase + (long)y * 256 + xbase + xi] =
        make_float2(tr[y * 17 + xi], ti[y * 17 + xi]);
  }
}

// FFT along z (length 512): LDS tile of 512(z) x 8(x) at fixed y; 1 col/wave.
__global__ void __launch_bounds__(256) k_pass_z(float2* __restrict__ buf,
                                                float sgn) {
  __shared__ float tr[512 * 9], ti[512 * 9];
  __shared__ float shr[8][272], shi[8][272];
  const int wave = threadIdx.x >> 5;
  const int y = blockIdx.x >> 5;
  const int xbase = (blockIdx.x & 31) * 8;

#pragma unroll
  for (int j = 0; j < 16; ++j) {
    const int e = threadIdx.x + 256 * j;  // 0..4095
    const int zz = e >> 3, xi = e & 7;
    const float2 v = buf[((long)zz * 256 + y) * 256 + xbase + xi];
    tr[zz * 9 + xi] = v.x;
    ti[zz * 9 + xi] = v.y;
  }
  __syncthreads();
  fft512_wave(tr + wave, ti + wave, 9, shr[wave], shi[wave], sgn);
  __syncthreads();
  // DIF output: even k in rows 0..255, odd k in rows 256..511.
#pragma unroll
  for (int j = 0; j < 16; ++j) {
    const int e = threadIdx.x + 256 * j;
    const int zz = e >> 3, xi = e & 7;
    const int src = (zz & 1) * 256 + (zz >> 1);
    buf[((long)zz * 256 + y) * 256 + xbase + xi] =
        make_float2(tr[src * 9 + xi], ti[src * 9 + xi]);
  }
}

// ---------------------------------------------------------------------------
// Stolt resampling: fuses fftshift, trilinear gather, z<=M zeroing,
// |gz|/gznew scaling, and ifftshift (both rolls are half-period shifts, so
// +256 mod 512 / +128 mod 256 implement them in either direction). A -> B.
// ---------------------------------------------------------------------------
__global__ void __launch_bounds__(256) k_stolt(const float2* __restrict__ A,
                                               float2* __restrict__ B) {
  const long idx = (long)blockIdx.x * 256 + threadIdx.x;  // < 33,554,432
  const int px = (int)(idx & 255);
  const int py = (int)((idx >> 8) & 255);
  const int pz = (int)(idx >> 16);
  float vr = 0.f, vi = 0.f;
  if (pz > 256) {  // t[:, :M+1] = 0
    const float gx = (float)(px - 128) * (1.0f / 128.0f);
    const float gy = (float)(py - 128) * (1.0f / 128.0f);
    const float gz = (float)(pz - 256) * (1.0f / 256.0f);
    const float fkrange = 0.1024f;  // (N*trange/(M*width*4))^2
    const float gznew = sqrtf(fkrange * (gx * gx + gy * gy) + gz * gz);
    // grid_sample, align_corners=False: ix = ((g+1)*W - 1)/2
    const float ix = ((gx + 1.f) * 256.f - 1.f) * 0.5f;
    const float iy = ((gy + 1.f) * 256.f - 1.f) * 0.5f;
    const float iz = ((gznew + 1.f) * 512.f - 1.f) * 0.5f;
    const float fx0 = floorf(ix), fy0 = floorf(iy), fz0 = floorf(iz);
    const float fx = ix - fx0, fy = iy - fy0, fz = iz - fz0;
    const int x0 = (int)fx0, y0 = (int)fy0, z0 = (int)fz0;
#pragma unroll
    for (int dz = 0; dz < 2; ++dz) {
#pragma unroll
      for (int dy = 0; dy < 2; ++dy) {
#pragma unroll
        for (int dx = 0; dx < 2; ++dx) {
          const int xc = x0 + dx, yc = y0 + dy, zc = z0 + dz;
          if (xc < 0 || xc >= 256 || yc < 0 || yc >= 256 || zc < 0 || zc >= 512)
            continue;
          const float w = (dx ? fx : 1.f - fx) * (dy ? fy : 1.f - fy) *
                          (dz ? fz : 1.f - fz);
          const float2 s =
              A[(((long)((zc + 256) & 511)) * 256 + ((yc + 128) & 255)) * 256 +
                ((xc + 128) & 255)];
          vr += s.x * w;
          vi += s.y * w;
        }
      }
    }
    const float scl = fabsf(gz) / (gznew + 1e-8f);
    vr *= scl;
    vi *= scl;
  }
  const long q = (((long)((pz + 256) & 511)) * 256 + ((py + 128) & 255)) * 256 +
                 ((px + 128) & 255);
  B[q] = make_float2(vr, vi);
}

// Squared magnitude of the [256,128,128] corner, with ifft 1/N^2 folded in.
__global__ void __launch_bounds__(256) k_out(const float2* __restrict__ B,
                                             float* __restrict__ out) {
  const long idx = (long)blockIdx.x * 256 + threadIdx.x;
  if (idx >= 4194304L) return;
  const int w = (int)(idx & 127);
  const int h = (int)((idx >> 7) & 127);
  const int t = (int)(idx >> 14);
  const float2 v = B[((long)t * 256 + h) * 256 + w];
  const float inv = 1.0f / 33554432.0f;  // 1/(512*256*256)
  out[idx] = (v.x * v.x + v.y * v.y) * (inv * inv);
}

// ---------------------------------------------------------------------------
extern "C" void kernel_launch(void* const* d_in, const int* in_sizes, int n_in,
                              void* d_out, int out_size, void* d_ws,
                              size_t ws_size, hipStream_t stream) {
  const float* feat = (const float*)d_in[0];
  const int* tbe = (const int*)d_in[1];
  const int* ten = (const int*)d_in[2];
  float2* A = (float2*)d_ws;                 // 512*256*256 complex = 268 MB
  float2* B = A + 33554432LL;                // second 268 MB buffer
  float* out = (float*)d_out;

  // Forward: fused pad/preprocess + FFT-x, then FFT-y, FFT-z.
  k_fwd_x<<<16384, 256, 0, stream>>>(feat, tbe, ten, A);
  k_pass_y<<<8192, 256, 0, stream>>>(A, -1.0f);
  k_pass_z<<<8192, 256, 0, stream>>>(A, -1.0f);
  // Stolt frequency resampling (fused shifts + mask + Jacobian).
  k_stolt<<<131072, 256, 0, stream>>>(A, B);
  // Inverse FFT (sign +1; normalization folded into k_out).
  k_pass_x<<<16384, 256, 0, stream>>>(B, 1.0f);
  k_pass_y<<<8192, 256, 0, stream>>>(B, 1.0f);
  k_pass_z<<<8192, 256, 0, stream>>>(B, 1.0f);
  // Squared magnitude of the corner volume.
  k_out<<<16384, 256, 0, stream>>>(B, out);
}